// TabularRescorlaWagnerPlusMinusValueUpdating_7670811590764
// MI455X (gfx1250) — compile-verified
//
#include <hip/hip_runtime.h>
#include <stdint.h>

// Rescorla-Wagner +/- value updating.
// N=8192 rows x T=1024 trials x K=10 options. Sequential over T per row.
// Memory-bound: ~335MB out + 64MB in => ~17us floor at 23.3 TB/s.
// Value table lives in VGPRs (cndmask gather/scatter); inputs are staged
// into LDS with gfx1250 async global->LDS loads, double buffered.

#define K_OPT    10
#define T_LEN    1024
#define RW_BLOCK 64
#define CH       32      // trials staged per chunk (32 ints = 128B = 8 x b128)
#define CHP      36      // padded dwords per lane-row in LDS (144B, 16B-aligned, gcd(36,64)=4 -> 2-way conflicts max)

#if defined(__HIP_DEVICE_COMPILE__) && defined(__gfx1250__) && \
    __has_builtin(__builtin_amdgcn_global_load_async_to_lds_b128) && \
    __has_builtin(__builtin_amdgcn_s_wait_asynccnt)
#define USE_ASYNC 1
#else
#define USE_ASYNC 0
#endif

typedef __attribute__((address_space(1))) void as1_void;
typedef __attribute__((address_space(3))) void as3_void;
typedef int v4i __attribute__((ext_vector_type(4)));
typedef __attribute__((address_space(1))) v4i as1_v4i;
typedef __attribute__((address_space(3))) v4i as3_v4i;

#if USE_ASYNC
// Recursively emit 8 pairs of b128 async loads. Offsets are applied to BOTH
// pointers explicitly (imm offset = 0) so the IR semantics are unambiguous;
// the backend may fold the common offset into INST_OFFSET (which the HW adds
// to both the LDS and global addresses, ISA 15.18).
template <int J>
struct AsyncB128 {
  static __device__ __forceinline__ void go(const int* cg, const float* rg,
                                            int* cl, float* rl) {
    AsyncB128<J - 1>::go(cg, rg, cl, rl);
    __builtin_amdgcn_global_load_async_to_lds_b128(
        (as1_v4i*)(as1_void*)const_cast<int*>(cg + J * 4),
        (as3_v4i*)(as3_void*)(cl + J * 4), 0, 0);
    __builtin_amdgcn_global_load_async_to_lds_b128(
        (as1_v4i*)(as1_void*)const_cast<float*>(rg + J * 4),
        (as3_v4i*)(as3_void*)(rl + J * 4), 0, 0);
  }
};
template <>
struct AsyncB128<-1> {
  static __device__ __forceinline__ void go(const int*, const float*, int*, float*) {}
};
#endif

// One RW step: emit v (pre-update) to `o`, then update v[c] in registers.
// Gather/scatter over the register file via cndmask chains (no LDS round trip
// in the serial dependence chain).
__device__ __forceinline__ void rw_step_store(float v[K_OPT], int c, float r,
                                              float ap, float am,
                                              float* __restrict__ o) {
  float chosen = v[0];
#pragma unroll
  for (int k = 1; k < K_OPT; ++k) chosen = (c == k) ? v[k] : chosen;

  float pe = r - chosen;
  pe = (r == r) ? pe : 0.0f;               // NaN reward -> zero prediction error
  float lr = (pe >= 0.0f) ? ap : am;
  float nv = __builtin_fmaf(lr, pe, chosen);

  // Emit pre-update values: b64 granularity (40B record is 8B-aligned);
  // the vectorizer merges adjacent trials' records into b128 stores.
#pragma unroll
  for (int j = 0; j < K_OPT / 2; ++j) {
    float2 p;
    p.x = v[2 * j];
    p.y = v[2 * j + 1];
    *(float2*)(o + 2 * j) = p;
  }

#pragma unroll
  for (int k = 0; k < K_OPT; ++k) v[k] = (c == k) ? nv : v[k];
}

__global__ __launch_bounds__(RW_BLOCK) void rw_scan_kernel(
    const int* __restrict__ choices, const float* __restrict__ rewards,
    const float* __restrict__ alpha_plus, const float* __restrict__ alpha_minus,
    const float* __restrict__ initial_values, float* __restrict__ out, int N) {
  const int tid = threadIdx.x;
  const int row = blockIdx.x * RW_BLOCK + tid;
  if (row >= N) return;

  const float ap = 1.0f / (1.0f + expf(-alpha_plus[0]));
  const float am = 1.0f / (1.0f + expf(-alpha_minus[0]));
  const float iv = 100.0f * tanhf(initial_values[0]);

  float v[K_OPT];
#pragma unroll
  for (int k = 0; k < K_OPT; ++k) v[k] = iv;

  const size_t inbase = (size_t)row * T_LEN;
  float* orow = out + (size_t)row * T_LEN * K_OPT;

#if USE_ASYNC
  __shared__ __align__(16) int   cbuf[2][RW_BLOCK][CHP];
  __shared__ __align__(16) float rbuf[2][RW_BLOCK][CHP];

  // Prime chunk 0 (16 async b128 ops outstanding).
  AsyncB128<7>::go(choices + inbase, rewards + inbase,
                   &cbuf[0][tid][0], &rbuf[0][tid][0]);

  const int nch = T_LEN / CH;
  for (int ch = 0; ch < nch; ++ch) {
    const int cur = ch & 1;
    if (ch + 1 < nch) {
      // Issue next chunk into the other buffer, then wait so that only the
      // 16 just-issued ops remain outstanding => current buffer is complete
      // (async loads complete in order, ISA 10.7/15.18).
      AsyncB128<7>::go(choices + inbase + (size_t)(ch + 1) * CH,
                       rewards + inbase + (size_t)(ch + 1) * CH,
                       &cbuf[cur ^ 1][tid][0], &rbuf[cur ^ 1][tid][0]);
      __builtin_amdgcn_s_wait_asynccnt(16);
    } else {
      __builtin_amdgcn_s_wait_asynccnt(0);
    }

    const int*   cb = &cbuf[cur][tid][0];
    const float* rb = &rbuf[cur][tid][0];
    float* o = orow + (size_t)ch * CH * K_OPT;
#pragma unroll 4
    for (int i = 0; i < CH; ++i) {
      rw_step_store(v, cb[i], rb[i], ap, am, o + (size_t)i * K_OPT);
    }
  }
#else
  // Fallback: direct loads with gfx1250 prefetch (global_prefetch_b8).
  for (int t = 0; t < T_LEN; ++t) {
    if (t + CH < T_LEN) {
      __builtin_prefetch(&choices[inbase + t + CH], 0, 1);
      __builtin_prefetch(&rewards[inbase + t + CH], 0, 1);
    }
    const int   c = choices[inbase + t];
    const float r = rewards[inbase + t];
    rw_step_store(v, c, r, ap, am, orow + (size_t)t * K_OPT);
  }
#endif
}

extern "C" void kernel_launch(void* const* d_in, const int* in_sizes, int n_in,
                              void* d_out, int out_size, void* d_ws, size_t ws_size,
                              hipStream_t stream) {
  const int*   choices        = (const int*)d_in[0];
  const float* rewards        = (const float*)d_in[1];
  const float* alpha_plus     = (const float*)d_in[2];
  const float* alpha_minus    = (const float*)d_in[3];
  const float* initial_values = (const float*)d_in[4];
  float* out = (float*)d_out;

  const int N = in_sizes[0] / T_LEN;  // 8192
  dim3 grid((N + RW_BLOCK - 1) / RW_BLOCK);
  dim3 block(RW_BLOCK);
  hipLaunchKernelGGL(rw_scan_kernel, grid, block, 0, stream,
                     choices, rewards, alpha_plus, alpha_minus, initial_values,
                     out, N);
}